// DLCF_DCA_71141838291760
// MI455X (gfx1250) — compile-verified
//
#include <hip/hip_runtime.h>

// ---------------------------------------------------------------------------
// DLCF/DCA masking kernel for MI455X (gfx1250).
// Memory-bound (AI ~0.18 flop/byte): read 100MB, write 300MB => ~18us floor
// at 23.3 TB/s. No matmul in the reference, so no WMMA; instead we use the
// CDNA5 async-tensor data path (global_load_async_to_lds_b128 + ASYNCcnt
// double-buffer pipeline), b128 vector accesses, and non-temporal stores for
// the write-once 300MB output stream (larger than the 192MB L2).
// ---------------------------------------------------------------------------

typedef float f4 __attribute__((ext_vector_type(4)));

#define S_LEN  256          // MAX_SEQ_LEN (fixed by reference)
#define TPB    256          // 8 wave32 per block
#define CHUNKS 12           // chunks per block in the streaming kernel
#define HV4    192          // H=768 floats per row = 192 float4

// ---------------------------------------------------------------------------
// Kernel 1: build per-(b,s) multiplier table. One block per batch.
//   mA = keep mask for depend_out, mB = keep mask for depended_out,
//   mC = scatter-priority multiplier for local_out2.
// ---------------------------------------------------------------------------
__global__ void build_masks_kernel(const int* __restrict__ dep,
                                   const int* __restrict__ dpd,
                                   const int* __restrict__ noc,
                                   const float* __restrict__ depw,
                                   const float* __restrict__ dpdw,
                                   f4* __restrict__ mult, int K) {
    __shared__ int fdep[S_LEN], fdpd[S_LEN], fnoc[S_LEN];
    const int b = blockIdx.x;
    const int t = threadIdx.x;
    fdep[t] = 0; fdpd[t] = 0; fnoc[t] = 0;
    __syncthreads();
    if (t < K) {
        int i = dep[b * K + t];
        if (i >= 0 && i + 1 < S_LEN) fdep[i + 1] = 1;
        i = dpd[b * K + t];
        if (i >= 0 && i + 1 < S_LEN) fdpd[i + 1] = 1;
        i = noc[b * K + t];
        if (i >= 0 && i + 1 < S_LEN) fnoc[i + 1] = 1;
    }
    __syncthreads();
    const float wA = depw[b];
    const float wB = dpdw[b];
    const float mA = (t == 0 || fdep[t]) ? 1.0f : 0.0f;
    const float mB = (t == 0 || fdpd[t]) ? 1.0f : 0.0f;
    float mC;
    if (t == 0)       mC = 1.0f;   // position 0 always copied
    else if (fnoc[t]) mC = 0.0f;   // no_connect zeroes last
    else if (fdpd[t]) mC = wB;     // depended overwrites depend
    else if (fdep[t]) mC = wA;
    else              mC = 0.0f;   // untouched positions stay zero
    f4 m; m.x = mA; m.y = mB; m.z = mC; m.w = 0.0f;
    mult[b * S_LEN + t] = m;
}

// ---------------------------------------------------------------------------
// Kernel 2: streaming scale. Each block owns CHUNKS contiguous 4KB chunks.
// Per-wave async double buffer: lane L async-copies its own 16B into LDS slot
// L of buffer (c+1)&1 while consuming buffer c&1. Since every lane reads back
// exactly the LDS bytes it loaded itself, s_wait_asynccnt (per-wave, in-order)
// is the only synchronization required -- no block barriers.
// ---------------------------------------------------------------------------
__global__ void __launch_bounds__(TPB)
scale_kernel(const f4* __restrict__ xv,
             const f4* __restrict__ mult,
             f4* __restrict__ o0, f4* __restrict__ o1, f4* __restrict__ o2,
             unsigned NV4) {
    __shared__ f4 tile[2][TPB];
    const unsigned tid  = threadIdx.x;
    const unsigned base = (unsigned)blockIdx.x * (TPB * CHUNKS);
    // Low 32 bits of the generic address of an LDS object == wave-relative
    // LDS byte offset (aperture lives in the high half) -> VDST operand.
    const unsigned lds0 = (unsigned)(unsigned long long)&tile[0][tid];
    const unsigned lds1 = (unsigned)(unsigned long long)&tile[1][tid];

    // Prologue: stage chunk 0.
    {
        const unsigned cv = base + tid;
        if (cv < NV4) {
            unsigned long long ga = (unsigned long long)(xv + cv);
            asm volatile("global_load_async_to_lds_b128 %0, %1, off"
                         :: "v"(lds0), "v"(ga) : "memory");
        }
    }

#pragma unroll
    for (int c = 0; c < CHUNKS; ++c) {
        if (c + 1 < CHUNKS) {
            // Issue next chunk, then wait for the current one (ASYNCcnt <= 1
            // leaves only the newest request in flight; async loads complete
            // in order within a wave).
            const unsigned nv = base + (unsigned)(c + 1) * TPB + tid;
            if (nv < NV4) {
                unsigned long long ga = (unsigned long long)(xv + nv);
                asm volatile("global_load_async_to_lds_b128 %0, %1, off"
                             :: "v"(((c + 1) & 1) ? lds1 : lds0), "v"(ga)
                             : "memory");
            }
            asm volatile("s_wait_asynccnt 0x1" ::: "memory");
        } else {
            asm volatile("s_wait_asynccnt 0x0" ::: "memory");
        }

        const unsigned cv = base + (unsigned)c * TPB + tid;
        if (cv < NV4) {
            f4 v = tile[c & 1][tid];          // ds_load_b128
            const unsigned row = cv / HV4;    // const-divide by 192
            f4 m = mult[row];                 // broadcast-friendly, keep in L1
            f4 r0 = v * m.x;
            f4 r1 = v * m.y;
            f4 r2 = v * m.z;
            // Output is streamed once and never re-read: NT hint keeps the
            // 300MB write stream from thrashing the 192MB L2.
            __builtin_nontemporal_store(r0, o0 + cv);
            __builtin_nontemporal_store(r1, o1 + cv);
            __builtin_nontemporal_store(r2, o2 + cv);
        }
    }
}

// ---------------------------------------------------------------------------
extern "C" void kernel_launch(void* const* d_in, const int* in_sizes, int n_in,
                              void* d_out, int out_size, void* d_ws, size_t ws_size,
                              hipStream_t stream) {
    const float* x    = (const float*)d_in[0];   // [B,S,H] f32
    const int*   dep  = (const int*)d_in[1];     // [B,K]
    const int*   dpd  = (const int*)d_in[2];     // [B,K]
    const int*   noc  = (const int*)d_in[3];     // [B,K]
    const float* depw = (const float*)d_in[4];   // [B]
    const float* dpdw = (const float*)d_in[5];   // [B]

    const int B = in_sizes[4];                   // 128
    const int K = in_sizes[1] / B;               // 64
    const long long N = (long long)in_sizes[0];  // B*S*H = 25,165,824

    float* out = (float*)d_out;                  // 3 outputs concatenated
    f4* mult = (f4*)d_ws;                        // B*S float4 = 512KB scratch

    build_masks_kernel<<<B, S_LEN, 0, stream>>>(dep, dpd, noc, depw, dpdw,
                                                mult, K);

    const unsigned NV4 = (unsigned)(N / 4);      // 6,291,456 float4
    const unsigned per_block = TPB * CHUNKS;     // 3072 float4 per block
    const int blocks = (int)((NV4 + per_block - 1) / per_block);  // 2048
    scale_kernel<<<blocks, TPB, 0, stream>>>(
        (const f4*)x, (const f4*)mult,
        (f4*)out, (f4*)(out + N), (f4*)(out + 2 * N), NV4);
}